// SlideTimeEncoder_70755291234328
// MI455X (gfx1250) — compile-verified
//
#include <hip/hip_runtime.h>
#include <hip/hip_bf16.h>
#include <stdint.h>

// SlideTimeEncoder: idx = clamp(floor(ts/1000)); out = W.T[idx] + b ; also pass ts through.
// Memory-bound (~84 MB @ 23.3 TB/s => ~3.6us floor). No GEMM -> no WMMA.
// gfx1250 path used: async global->LDS staging of the 32KB W table (ASYNCcnt),
// transposed on the fly so the hot loop is two ds_load_b128 per element.

#define B_DIM  4096
#define L_DIM  512
#define NTI    1000   // N_TIME_INTERVAL
#define ODIM   8      // OUT_DIM

__global__ __launch_bounds__(512)
void SlideTimeEncoder_kernel(const float* __restrict__ timestamp,
                             const float* __restrict__ W,
                             const float* __restrict__ bias,
                             float* __restrict__ out_emb,
                             float* __restrict__ out_ts)
{
    // Transposed table: wt[i*8 + d] = W[d*1000 + i]; 8000 floats = 32000 B of LDS.
    __shared__ __align__(16) float wt[NTI * ODIM];

    const int tid  = threadIdx.x;
    const int nthr = blockDim.x;

    // ---- Stage W (transposed) into LDS with gfx1250 async global->LDS DMA ----
    // Global side reads W linearly (contiguous per lane); LDS side scatters.
    for (int t = tid; t < NTI * ODIM; t += nthr) {
        const float* gp = W + t;            // t = d*NTI + i  (row-major W)
        int d = t / NTI;
        int i = t - d * NTI;
        // Low 32 bits of a generic pointer into LDS == LDS byte offset (shared aperture).
        unsigned lofs = (unsigned)(uintptr_t)(&wt[i * ODIM + d]);
        asm volatile("global_load_async_to_lds_b32 %0, %1, off"
                     :: "v"(lofs), "v"(gp)
                     : "memory");
    }
    asm volatile("s_wait_asynccnt 0" ::: "memory");
    __syncthreads();

    // Bias in VGPRs (broadcast load, L2/L0 hit).
    const float4 b0 = *(const float4*)(bias + 0);
    const float4 b1 = *(const float4*)(bias + 4);

    const long long NE     = (long long)B_DIM * L_DIM;        // 2,097,152
    const long long stride = (long long)gridDim.x * nthr;

    for (long long e = (long long)blockIdx.x * nthr + tid; e < NE; e += stride) {
        const long long r = e >> 9;          // row (L == 512)
        // timestamp row stride is L+1 = 513 => flat index = r*513 + c = e + r
        const float ts = timestamp[e + r];

        int idx = (int)floorf(ts / 1000.0f);
        idx = idx < 0 ? 0 : (idx > (NTI - 1) ? (NTI - 1) : idx);

        const float4* row = (const float4*)(&wt[idx * ODIM]);  // 32B-aligned LDS row
        const float4 w0 = row[0];
        const float4 w1 = row[1];

        float4 o0, o1;
        o0.x = w0.x + b0.x; o0.y = w0.y + b0.y; o0.z = w0.z + b0.z; o0.w = w0.w + b0.w;
        o1.x = w1.x + b1.x; o1.y = w1.y + b1.y; o1.z = w1.z + b1.z; o1.w = w1.w + b1.w;

        float4* op = (float4*)(out_emb + e * ODIM);  // 32B contiguous per lane: coalesced
        op[0] = o0;
        op[1] = o1;

        out_ts[e] = ts;
    }
}

extern "C" void kernel_launch(void* const* d_in, const int* in_sizes, int n_in,
                              void* d_out, int out_size, void* d_ws, size_t ws_size,
                              hipStream_t stream) {
    // setup_inputs order: input(int32, unused), timestamp(f32 B*(L+1)), W(f32 8*1000), b(f32 8)
    const float* timestamp = (const float*)d_in[1];
    const float* W         = (const float*)d_in[2];
    const float* bias      = (const float*)d_in[3];

    float* out_emb = (float*)d_out;                                        // (B, L, 8)
    float* out_ts  = (float*)d_out + (long long)B_DIM * L_DIM * ODIM;      // (B, L)

    dim3 grid(512);
    dim3 block(512);
    hipLaunchKernelGGL(SlideTimeEncoder_kernel, grid, block, 0, stream,
                       timestamp, W, bias, out_emb, out_ts);
}